// GraphCNNDiscriminator_58523224375723
// MI455X (gfx1250) — compile-verified
//
#include <hip/hip_runtime.h>
#include <hip/hip_bf16.h>

#define BATCH   32
#define NNODES  1024
#define CHEB    9
#define ROWS_TOT (BATCH * NNODES)   // 32768
#define WROWS   8

typedef float v2f __attribute__((ext_vector_type(2)));
typedef float v8f __attribute__((ext_vector_type(8)));

// ---------------------------------------------------------------------------
// Laplacian stage 1: W[b,i,j] = A[b,i,j] / (||x[b,i]-x[b,j]|| + 1e-3)
// ---------------------------------------------------------------------------
__global__ __launch_bounds__(256)
void w_pair_kernel(const float* __restrict__ x, const float* __restrict__ A,
                   float* __restrict__ W) {
  int b  = blockIdx.y;
  int i0 = blockIdx.x * WROWS;
  __shared__ float xs[NNODES * 3];
  const float* xb = x + (size_t)b * NNODES * 3;
  for (int t = threadIdx.x; t < NNODES * 3; t += blockDim.x) xs[t] = xb[t];
  __syncthreads();
  for (int ii = 0; ii < WROWS; ++ii) {
    int i = i0 + ii;
    float xi0 = xs[i * 3 + 0], xi1 = xs[i * 3 + 1], xi2 = xs[i * 3 + 2];
    const float* Ar = A + ((size_t)b * NNODES + i) * NNODES;
    float*       Wr = W + ((size_t)b * NNODES + i) * NNODES;
    for (int j = threadIdx.x; j < NNODES; j += blockDim.x) {
      float d0 = xi0 - xs[j * 3 + 0];
      float d1 = xi1 - xs[j * 3 + 1];
      float d2 = xi2 - xs[j * 3 + 2];
      float dist = sqrtf(d0 * d0 + d1 * d1 + d2 * d2);
      Wr[j] = Ar[j] / (dist + 0.001f);
    }
  }
}

// Laplacian stage 2: d[b,j] = 1/sqrt(sum_i W[b,i,j])  (column sums, coalesced)
__global__ __launch_bounds__(256)
void d_col_kernel(const float* __restrict__ W, float* __restrict__ dvec) {
  int b = blockIdx.y;
  int j = blockIdx.x * 256 + threadIdx.x;
  const float* Wb = W + (size_t)b * NNODES * NNODES;
  float s = 0.f;
  for (int i = 0; i < NNODES; ++i) s += Wb[(size_t)i * NNODES + j];
  dvec[b * NNODES + j] = 1.0f / sqrtf(s);
}

// Laplacian stage 3 (in place): L = -0.5*I - 0.5 * d_i * W_ij * d_j
__global__ __launch_bounds__(256)
void l_scale_kernel(float* __restrict__ W, const float* __restrict__ dvec) {
  int b  = blockIdx.y;
  int i0 = blockIdx.x * WROWS;
  __shared__ float ds[NNODES];
  const float* db = dvec + b * NNODES;
  for (int t = threadIdx.x; t < NNODES; t += blockDim.x) ds[t] = db[t];
  __syncthreads();
  for (int ii = 0; ii < WROWS; ++ii) {
    int i = i0 + ii;
    float di = ds[i];
    float* Wr = W + ((size_t)b * NNODES + i) * NNODES;
    for (int j = threadIdx.x; j < NNODES; j += blockDim.x) {
      float v = -0.5f * di * ds[j] * Wr[j];
      if (j == i) v -= 0.5f;
      Wr[j] = v;
    }
  }
}

// Zero-pad x [B,N,3] -> TX [B,N,16] so layer-1 GEMMs have no divergent guards.
__global__ __launch_bounds__(256)
void pad_x_kernel(const float* __restrict__ x, float* __restrict__ TX) {
  int idx = blockIdx.x * 256 + threadIdx.x;   // over B*N*16
  int rc  = idx >> 4;
  int c   = idx & 15;
  TX[idx] = (c < 3) ? x[rc * 3 + c] : 0.f;
}

// ---------------------------------------------------------------------------
// Chebyshev recursion GEMM via V_WMMA_F32_16X16X4_F32 (full fp32 precision).
//   recur==0 : Tout = L @ Tin
//   recur==1 : Tout = 2*(L @ Tin) - Tprev
// F must be a multiple of 16 (layer-1 data is zero-padded to 16).
// Block = 4 waves, 64 rows of L. The 64xF B-chunk of T is contiguous in
// memory -> staged into LDS with coalesced float4 copies and shared by all
// 4 waves; row stride padded (+8) so half-wave ds reads hit disjoint banks.
// Fragment layouts per ISA 7.12.2:
//   A 16x4 f32 : lane m=lane&15, kh=lane>>4; v0=K(2kh), v1=K(2kh+1)
//   B  4x16    : lane col=lane&15, kh=lane>>4; v0=row(2kh), v1=row(2kh+1)
//   C/D 16x16  : vgpr v -> row m0+v+8*(lane>>4), col n0+(lane&15)
// ---------------------------------------------------------------------------
template <int F>
__global__ __launch_bounds__(128)
void cheb_gemm(const float* __restrict__ Lm, const float* __restrict__ Tin,
               const float* __restrict__ Tprev, float* __restrict__ Tout,
               int recur) {
  static_assert(F % 16 == 0, "F must be a multiple of 16");
  constexpr int NT   = F / 16;
  constexpr int KC   = 64;        // k-chunk staged in LDS
  constexpr int BSTR = F + 8;     // padded LDS row stride (banks)
  __shared__ float Bs[KC * BSTR];

  int b    = blockIdx.y;
  int wave = threadIdx.x >> 5;
  int lane = threadIdx.x & 31;
  int m0   = (blockIdx.x * 4 + wave) << 4;
  int mrow = lane & 15;
  int kh   = lane >> 4;
  int col  = lane & 15;

  const float* Arow = Lm + ((size_t)b * NNODES + (m0 + mrow)) * NNODES;
  const float* Tb   = Tin + (size_t)b * NNODES * F;

  v8f acc[NT];
#pragma unroll
  for (int t = 0; t < NT; ++t)
    acc[t] = (v8f){0.f, 0.f, 0.f, 0.f, 0.f, 0.f, 0.f, 0.f};

  for (int kc = 0; kc < NNODES; kc += KC) {
    // Cooperative contiguous copy of T[kc:kc+KC, 0:F] into LDS (float4).
    const float4* src = (const float4*)(Tb + (size_t)kc * F);
#pragma unroll
    for (int t = threadIdx.x; t < KC * F / 4; t += 128) {
      float4 v4 = src[t];
      int r = (t * 4) / F;
      int c = (t * 4) % F;
      *(float4*)&Bs[r * BSTR + c] = v4;
    }
    __syncthreads();

#pragma unroll
    for (int k0 = 0; k0 < KC; k0 += 4) {
      v2f a = *(const v2f*)(Arow + kc + k0 + 2 * kh);   // 8B aligned
      int krl = k0 + 2 * kh;
#pragma unroll
      for (int nt = 0; nt < NT; ++nt) {
        int n = nt * 16 + col;
        v2f bf;
        bf.x = Bs[krl * BSTR + n];
        bf.y = Bs[(krl + 1) * BSTR + n];
        acc[nt] = __builtin_amdgcn_wmma_f32_16x16x4_f32(
            false, a, false, bf, (short)0, acc[nt], false, false);
      }
    }
    __syncthreads();
  }

  float*       Ob = Tout + (size_t)b * NNODES * F;
  const float* Pb = recur ? (Tprev + (size_t)b * NNODES * F) : nullptr;
#pragma unroll
  for (int nt = 0; nt < NT; ++nt) {
    int n = nt * 16 + col;
#pragma unroll
    for (int v = 0; v < 8; ++v) {
      int r = m0 + v + 8 * kh;
      float val = acc[nt][v];
      if (recur) val = 2.f * val - Pb[(size_t)r * F + n];
      Ob[(size_t)r * F + n] = val;
    }
  }
}

// ---------------------------------------------------------------------------
// Projection accumulate: acc[r,fo] (+)= T[r,:FIN] @ W[krow0:krow0+FIN, :FOUT]
// T rows have stride TS (16 for padded layer-1 tensors, else FIN).
// ---------------------------------------------------------------------------
template <int FIN, int FOUT, int TS>
__global__ __launch_bounds__(256)
void proj_kernel(const float* __restrict__ T, const float* __restrict__ Wfull,
                 int krow0, const float* __restrict__ bias,
                 float* __restrict__ acc, int init) {
  __shared__ float Ws[FIN * FOUT];
  for (int t = threadIdx.x; t < FIN * FOUT; t += blockDim.x)
    Ws[t] = Wfull[(size_t)(krow0 + t / FOUT) * FOUT + (t % FOUT)];
  __syncthreads();
  int idx = blockIdx.x * blockDim.x + threadIdx.x;   // over ROWS_TOT*FOUT
  int r   = idx / FOUT;
  int fo  = idx & (FOUT - 1);
  const float* Tr = T + (size_t)r * TS;
  float s = init ? bias[fo] : acc[idx];
#pragma unroll
  for (int f = 0; f < FIN; ++f) s = fmaf(Tr[f], Ws[f * FOUT + fo], s);
  acc[idx] = s;
}

// ---------------------------------------------------------------------------
// BatchNorm (train, biased var) over 32768 rows, then ELU, in place.
// ---------------------------------------------------------------------------
__global__ void zero_kernel(float* __restrict__ p, int n) {
  int i = blockIdx.x * blockDim.x + threadIdx.x;
  if (i < n) p[i] = 0.f;
}

template <int C>
__global__ __launch_bounds__(256)
void bn_stats(const float* __restrict__ acc, float* __restrict__ sums) {
  constexpr int SUB = 256 / C;
  int c   = threadIdx.x & (C - 1);
  int sub = threadIdx.x / C;
  float s = 0.f, q = 0.f;
  for (int r = blockIdx.x * SUB + sub; r < ROWS_TOT; r += gridDim.x * SUB) {
    float v = acc[(size_t)r * C + c];
    s += v; q += v * v;
  }
  __shared__ float sh[256], shq[256];
  sh[threadIdx.x] = s; shq[threadIdx.x] = q;
  __syncthreads();
  for (int st = SUB >> 1; st > 0; st >>= 1) {
    if (sub < st) {
      sh[sub * C + c]  += sh[(sub + st) * C + c];
      shq[sub * C + c] += shq[(sub + st) * C + c];
    }
    __syncthreads();
  }
  if (sub == 0) {
    atomicAdd(&sums[c],     sh[c]);
    atomicAdd(&sums[C + c], shq[c]);
  }
}

template <int C>
__global__ __launch_bounds__(256)
void bn_norm_elu(float* __restrict__ acc, const float* __restrict__ sums,
                 const float* __restrict__ gamma, const float* __restrict__ beta) {
  int idx = blockIdx.x * blockDim.x + threadIdx.x;
  int c   = idx & (C - 1);
  const float invR = 1.0f / (float)ROWS_TOT;
  float mu  = sums[c] * invR;
  float var = sums[C + c] * invR - mu * mu;
  float inv = 1.0f / sqrtf(var + 1e-5f);
  float v = (acc[idx] - mu) * inv * gamma[c] + beta[c];
  acc[idx] = v > 0.f ? v : (expf(v) - 1.0f);
}

// ---------------------------------------------------------------------------
// Mean over N then FC [64]->[1]
// ---------------------------------------------------------------------------
__global__ __launch_bounds__(256)
void final_kernel(const float* __restrict__ H, const float* __restrict__ Wfc,
                  const float* __restrict__ bfc, float* __restrict__ out) {
  int b   = blockIdx.x;
  int c   = threadIdx.x & 63;
  int sub = threadIdx.x >> 6;   // 0..3
  const float* Hb = H + (size_t)b * NNODES * 64;
  float s = 0.f;
  for (int n = sub; n < NNODES; n += 4) s += Hb[(size_t)n * 64 + c];
  __shared__ float sh[256];
  sh[threadIdx.x] = s;
  __syncthreads();
  if (sub == 0) {
    float t = sh[c] + sh[64 + c] + sh[128 + c] + sh[192 + c];
    sh[c] = t * (1.0f / (float)NNODES) * Wfc[c];
  }
  __syncthreads();
  if (threadIdx.x == 0) {
    float r = 0.f;
    for (int k = 0; k < 64; ++k) r += sh[k];
    out[b] = r + bfc[0];
  }
}

// ---------------------------------------------------------------------------
// Host-side layer driver.  TS = row stride of the T buffers (>= FIN, mult 16)
// ---------------------------------------------------------------------------
template <int FIN, int FOUT, int TS>
static void run_layer(const float* Hin, const float* Wl, const float* bl,
                      const float* gl, const float* bel, const float* Lbuf,
                      float* T0, float* T1, float* T2, float* acc, float* sums,
                      hipStream_t stream) {
  dim3 ggrid(NNODES / 64, BATCH);
  int projBlocks = ROWS_TOT * FOUT / 256;

  // k=0 term (starts from bias)
  proj_kernel<FIN, FOUT, TS><<<projBlocks, 256, 0, stream>>>(Hin, Wl, 0, bl, acc, 1);
  // T1 = L @ T0
  cheb_gemm<TS><<<ggrid, 128, 0, stream>>>(Lbuf, Hin, nullptr, T0, 0);
  proj_kernel<FIN, FOUT, TS><<<projBlocks, 256, 0, stream>>>(T0, Wl, FIN, bl, acc, 0);

  float* bufs[3] = {T0, T1, T2};
  const float* Tpp = Hin;   // never written (may be the padded input)
  const float* Tp  = T0;
  for (int k = 2; k < CHEB; ++k) {
    float* Tn = bufs[(k - 1) % 3];
    cheb_gemm<TS><<<ggrid, 128, 0, stream>>>(Lbuf, Tp, Tpp, Tn, 1);
    proj_kernel<FIN, FOUT, TS><<<projBlocks, 256, 0, stream>>>(Tn, Wl, k * FIN, bl, acc, 0);
    Tpp = Tp; Tp = Tn;
  }

  // BatchNorm + ELU in place on acc
  zero_kernel<<<1, 256, 0, stream>>>(sums, 2 * FOUT);
  bn_stats<FOUT><<<64, 256, 0, stream>>>(acc, sums);
  bn_norm_elu<FOUT><<<ROWS_TOT * FOUT / 256, 256, 0, stream>>>(acc, sums, gl, bel);
}

extern "C" void kernel_launch(void* const* d_in, const int* in_sizes, int n_in,
                              void* d_out, int out_size, void* d_ws, size_t ws_size,
                              hipStream_t stream) {
  (void)in_sizes; (void)n_in; (void)out_size; (void)ws_size;
  const float* x   = (const float*)d_in[0];
  const float* A   = (const float*)d_in[1];
  const float* W1  = (const float*)d_in[2];
  const float* b1  = (const float*)d_in[3];
  const float* g1  = (const float*)d_in[4];
  const float* be1 = (const float*)d_in[5];
  const float* W2  = (const float*)d_in[6];
  const float* b2  = (const float*)d_in[7];
  const float* g2  = (const float*)d_in[8];
  const float* be2 = (const float*)d_in[9];
  const float* W3  = (const float*)d_in[10];
  const float* b3  = (const float*)d_in[11];
  const float* g3  = (const float*)d_in[12];
  const float* be3 = (const float*)d_in[13];
  const float* Wfc = (const float*)d_in[14];
  const float* bfc = (const float*)d_in[15];

  // Workspace layout (floats); total ~178 MB
  float* ws   = (float*)d_ws;
  float* Lbuf = ws;                                           // 32*1024*1024
  float* dvec = Lbuf + (size_t)BATCH * NNODES * NNODES;       // 32*1024
  float* sums = dvec + (size_t)BATCH * NNODES;                // 256 (2*64 used)
  const size_t TSZ = (size_t)BATCH * NNODES * 64;
  float* TA  = sums + 256;
  float* TB  = TA + TSZ;
  float* TC  = TB + TSZ;
  float* AC1 = TC + TSZ;
  float* AC2 = AC1 + TSZ;
  float* TX  = AC2 + TSZ;    // padded x, [32,1024,16]

  // Laplacian (stored into Lbuf; W computed in place then scaled)
  w_pair_kernel<<<dim3(NNODES / WROWS, BATCH), 256, 0, stream>>>(x, A, Lbuf);
  d_col_kernel<<<dim3(NNODES / 256, BATCH), 256, 0, stream>>>(Lbuf, dvec);
  l_scale_kernel<<<dim3(NNODES / WROWS, BATCH), 256, 0, stream>>>(Lbuf, dvec);

  // Pad x to 16 channels for divergence-free WMMA GEMMs
  pad_x_kernel<<<(BATCH * NNODES * 16) / 256, 256, 0, stream>>>(x, TX);

  // Three Chebyshev conv + BN + ELU layers
  run_layer<3, 32, 16>(TX,  W1, b1, g1, be1, Lbuf, TA, TB, TC, AC1, sums, stream);
  run_layer<32, 64, 32>(AC1, W2, b2, g2, be2, Lbuf, TA, TB, TC, AC2, sums, stream);
  run_layer<64, 64, 64>(AC2, W3, b3, g3, be3, Lbuf, TA, TB, TC, AC1, sums, stream);

  // Mean over nodes + final FC
  final_kernel<<<BATCH, 256, 0, stream>>>(AC1, Wfc, bfc, (float*)d_out);
}